// TextureLoss_18141941858429
// MI455X (gfx1250) — compile-verified
//
#include <hip/hip_runtime.h>
#include <math.h>

typedef __attribute__((ext_vector_type(2))) float v2f;
typedef __attribute__((ext_vector_type(8))) float v8f;

#define C_DIM 512
#define D_DIM 65536
#define BLK 64                       // macro output tile (BLK x BLK)
#define NT (C_DIM / BLK)             // 8 macro tiles per side
#define NPAIRS (NT * (NT + 1) / 2)   // 36 upper-triangular macro tiles
#define TEAMS 4                      // K-split teams inside a workgroup
#define THREADS 512                  // 16 waves = 4 teams x 4 compute waves
#define KSEG (D_DIM / TEAMS)         // 16384 K per team
#define KC 8                         // K chunk staged in LDS per team
#define LSTR 12                      // padded LDS row stride (48B: 16B aligned, bank-conflict free)
#define PLANE (BLK * LSTR)           // 768 floats per plane
#define NCHUNK (KSEG / KC)           // 2048 chunks per team (even -> exact 2x unroll)

__device__ __forceinline__ v8f wmma_f32(v2f a, v2f b, v8f c) {
  return __builtin_amdgcn_wmma_f32_16x16x4_f32(
      false, a, false, b, (short)0, c, false, false);
}

// CDNA5 async copy: 16B global -> LDS per lane, tracked by ASYNCcnt.
__device__ __forceinline__ void async_copy_b128(unsigned lds_addr, const void* gptr) {
  asm volatile("global_load_async_to_lds_b128 %0, %1, off"
               :: "v"(lds_addr), "v"(gptr) : "memory");
}
__device__ __forceinline__ void wait_async0() {
  asm volatile("s_wait_asynccnt 0" ::: "memory");
}

__device__ __forceinline__ void stage4(unsigned d0, unsigned d1, unsigned d2, unsigned d3,
                                       const float* pAa, const float* pBa,
                                       const float* pAt, const float* pBt, size_t k) {
  async_copy_b128(d0, pAa + k);
  async_copy_b128(d1, pBa + k);
  async_copy_b128(d2, pAt + k);
  async_copy_b128(d3, pBt + k);
}

__device__ __forceinline__ void compute_chunk(const float* P,
    int rel_a0, int rel_a1, int rel_b0, int rel_b1,
    v8f& aa00, v8f& aa01, v8f& aa10, v8f& aa11,
    v8f& at00, v8f& at01, v8f& at10, v8f& at11) {
#pragma unroll
  for (int ks = 0; ks < KC; ks += 4) {
    v2f aA0 = *(const v2f*)(P + 0 * PLANE + rel_a0 + ks);
    v2f aA1 = *(const v2f*)(P + 0 * PLANE + rel_a1 + ks);
    v2f bA0 = *(const v2f*)(P + 1 * PLANE + rel_b0 + ks);
    v2f bA1 = *(const v2f*)(P + 1 * PLANE + rel_b1 + ks);
    v2f aT0 = *(const v2f*)(P + 2 * PLANE + rel_a0 + ks);
    v2f aT1 = *(const v2f*)(P + 2 * PLANE + rel_a1 + ks);
    v2f bT0 = *(const v2f*)(P + 3 * PLANE + rel_b0 + ks);
    v2f bT1 = *(const v2f*)(P + 3 * PLANE + rel_b1 + ks);

    aa00 = wmma_f32(aA0, bA0, aa00);
    aa01 = wmma_f32(aA0, bA1, aa01);
    aa10 = wmma_f32(aA1, bA0, aa10);
    aa11 = wmma_f32(aA1, bA1, aa11);
    at00 = wmma_f32(aT0, bT0, at00);
    at01 = wmma_f32(aT0, bT1, at01);
    at10 = wmma_f32(aT1, bT0, at10);
    at11 = wmma_f32(aT1, bT1, at11);
  }
}

__device__ __forceinline__ void st8(float* p, v8f v) {
#pragma unroll
  for (int i = 0; i < 8; ++i) p[i] = v[i];
}
__device__ __forceinline__ void add8(v8f& a, const float* p) {
#pragma unroll
  for (int i = 0; i < 8; ++i) a[i] += p[i];
}

__global__ void __launch_bounds__(THREADS)
gram_loss_kernel(const float* __restrict__ a0, const float* __restrict__ a1,
                 const float* __restrict__ a2, const float* __restrict__ a3,
                 const float* __restrict__ t0, const float* __restrict__ t1,
                 const float* __restrict__ t2, const float* __restrict__ t3,
                 float* __restrict__ ws) {
  // [team][buffer][plane][row*LSTR + k]; planes: 0=act rows-i, 1=act rows-j, 2=tgt rows-i, 3=tgt rows-j
  // 4*2*4*768 floats = 96KB; reused afterwards as the cross-team reduction buffer.
  __shared__ __align__(16) float smem[TEAMS * 2 * 4 * PLANE];
  __shared__ float rnum[4], rden[4];

  const int layer = blockIdx.y;
  const float* act = (layer == 0) ? a0 : (layer == 1) ? a1 : (layer == 2) ? a2 : a3;
  const float* tgt = (layer == 0) ? t0 : (layer == 1) ? t1 : (layer == 2) ? t2 : t3;

  // linear -> upper-triangular (ti, tj), ti <= tj
  int rem = blockIdx.x;
  int ti = 0;
  while (rem >= NT - ti) { rem -= NT - ti; ++ti; }
  const int tj = ti + rem;
  const int i0 = ti * BLK;
  const int j0 = tj * BLK;

  const int tid  = threadIdx.x;
  const int lane = tid & 31;
  const int wave = tid >> 5;     // 0..15
  const int team = wave >> 2;    // 0..3 : K segment
  const int cw   = wave & 3;     // 0..3 : compute wave in team
  const int wr   = cw >> 1;      // 0..1 : 32-row strip
  const int wc   = cw & 1;       // 0..1 : 32-col strip

  // ---- async staging mapping: thread-in-team lt covers (row, half) -> one b128 per plane per chunk
  const int lt   = tid & 127;
  const int row  = lt >> 1;        // 0..63
  const int half = lt & 1;         // 0..1  -> k offset 0 or 4

  const size_t kbase = (size_t)team * KSEG + half * 4;
  const float* pAa = act + (size_t)(i0 + row) * D_DIM + kbase;
  const float* pBa = act + (size_t)(j0 + row) * D_DIM + kbase;
  const float* pAt = tgt + (size_t)(i0 + row) * D_DIM + kbase;
  const float* pBt = tgt + (size_t)(j0 + row) * D_DIM + kbase;

  const unsigned smem_base = (unsigned)(uintptr_t)(void*)&smem[0];
  const unsigned st_off    = (unsigned)((row * LSTR + half * 4) * 4);
  const unsigned team_off  = (unsigned)(((team * 2) * 4 * PLANE) * 4);
  // loop-invariant LDS byte addresses, compile-time buffer/plane selection (no movrels)
  const unsigned d0p0 = smem_base + team_off + 0 * PLANE * 4 + st_off;
  const unsigned d0p1 = smem_base + team_off + 1 * PLANE * 4 + st_off;
  const unsigned d0p2 = smem_base + team_off + 2 * PLANE * 4 + st_off;
  const unsigned d0p3 = smem_base + team_off + 3 * PLANE * 4 + st_off;
  const unsigned d1p0 = d0p0 + 4 * PLANE * 4;
  const unsigned d1p1 = d0p1 + 4 * PLANE * 4;
  const unsigned d1p2 = d0p2 + 4 * PLANE * 4;
  const unsigned d1p3 = d0p3 + 4 * PLANE * 4;

  // ---- WMMA fragment offsets (floats, plane-relative); ISA 16x4 fp32 A layout, B mirrored
  const int fm = lane & 15;
  const int fk = (lane >> 4) * 2;
  const int rel_a0 = (wr * 32 + fm) * LSTR + fk;
  const int rel_a1 = rel_a0 + 16 * LSTR;
  const int rel_b0 = (wc * 32 + fm) * LSTR + fk;
  const int rel_b1 = rel_b0 + 16 * LSTR;

  const float* P0 = &smem[(size_t)((team * 2 + 0) * 4) * PLANE];
  const float* P1 = &smem[(size_t)((team * 2 + 1) * 4) * PLANE];

  v8f aa00 = {}, aa01 = {}, aa10 = {}, aa11 = {};
  v8f at00 = {}, at01 = {}, at10 = {}, at11 = {};

  // prologue: stage chunk 0 into buffer 0
  stage4(d0p0, d0p1, d0p2, d0p3, pAa, pBa, pAt, pBt, 0);
  wait_async0();
  __syncthreads();

  for (int c = 0; c < NCHUNK; c += 2) {
    // ping: prefetch chunk c+1 into buf1, compute buf0
    stage4(d1p0, d1p1, d1p2, d1p3, pAa, pBa, pAt, pBt, (size_t)(c + 1) * KC);
    compute_chunk(P0, rel_a0, rel_a1, rel_b0, rel_b1,
                  aa00, aa01, aa10, aa11, at00, at01, at10, at11);
    wait_async0();
    __syncthreads();

    // pong: prefetch chunk c+2 into buf0, compute buf1
    if (c + 2 < NCHUNK) {
      stage4(d0p0, d0p1, d0p2, d0p3, pAa, pBa, pAt, pBt, (size_t)(c + 2) * KC);
    }
    compute_chunk(P1, rel_a0, rel_a1, rel_b0, rel_b1,
                  aa00, aa01, aa10, aa11, at00, at01, at10, at11);
    wait_async0();
    __syncthreads();
  }

  // ---- cross-team partial-Gram reduction through LDS (must happen before the nonlinear norm)
  if (team != 0) {
    float* dst = smem + (size_t)(((team - 1) * 4 + cw) * 32 + lane) * 64;
    st8(dst +  0, aa00); st8(dst +  8, aa01); st8(dst + 16, aa10); st8(dst + 24, aa11);
    st8(dst + 32, at00); st8(dst + 40, at01); st8(dst + 48, at10); st8(dst + 56, at11);
  }
  __syncthreads();

  if (team == 0) {
#pragma unroll
    for (int s = 0; s < 3; ++s) {
      const float* src = smem + (size_t)((s * 4 + cw) * 32 + lane) * 64;
      add8(aa00, src +  0); add8(aa01, src +  8); add8(aa10, src + 16); add8(aa11, src + 24);
      add8(at00, src + 32); add8(at01, src + 40); add8(at10, src + 48); add8(at11, src + 56);
    }

    // fused Frobenius terms: num = sum (Gt-Ga)^2, den = sum Gt^2
    float num = 0.0f, den = 0.0f;
#pragma unroll
    for (int i = 0; i < 8; ++i) {
      float d;
      d = at00[i] - aa00[i]; num += d * d; den += at00[i] * at00[i];
      d = at01[i] - aa01[i]; num += d * d; den += at01[i] * at01[i];
      d = at10[i] - aa10[i]; num += d * d; den += at10[i] * at10[i];
      d = at11[i] - aa11[i]; num += d * d; den += at11[i] * at11[i];
    }
    const float w = (ti == tj) ? 1.0f : 2.0f;  // symmetry weight
    num *= w;
    den *= w;

#pragma unroll
    for (int off = 16; off > 0; off >>= 1) {
      num += __shfl_down(num, off);
      den += __shfl_down(den, off);
    }
    if (lane == 0) { rnum[cw] = num; rden[cw] = den; }
  }
  __syncthreads();

  if (tid == 0) {
    float n = rnum[0] + rnum[1] + rnum[2] + rnum[3];
    float d = rden[0] + rden[1] + rden[2] + rden[3];
    atomicAdd(&ws[layer], n);
    atomicAdd(&ws[4 + layer], d);
  }
}

__global__ void init_ws_kernel(float* __restrict__ ws) {
  if (threadIdx.x < 8) ws[threadIdx.x] = 0.0f;
}

__global__ void finalize_kernel(const float* __restrict__ ws, float* __restrict__ out) {
  if (threadIdx.x == 0) {
    float s = 0.0f;
#pragma unroll
    for (int l = 0; l < 4; ++l) s += sqrtf(ws[l]) / sqrtf(ws[4 + l]);
    out[0] = s;
  }
}

extern "C" void kernel_launch(void* const* d_in, const int* in_sizes, int n_in,
                              void* d_out, int out_size, void* d_ws, size_t ws_size,
                              hipStream_t stream) {
  const float* a0 = (const float*)d_in[0];
  const float* a1 = (const float*)d_in[1];
  const float* a2 = (const float*)d_in[2];
  const float* a3 = (const float*)d_in[3];
  const float* t0 = (const float*)d_in[4];
  const float* t1 = (const float*)d_in[5];
  const float* t2 = (const float*)d_in[6];
  const float* t3 = (const float*)d_in[7];
  float* ws  = (float*)d_ws;
  float* out = (float*)d_out;

  init_ws_kernel<<<1, 32, 0, stream>>>(ws);

  dim3 grid(NPAIRS, 4);  // 36 upper-tri macro tiles x 4 layers
  gram_loss_kernel<<<grid, THREADS, 0, stream>>>(a0, a1, a2, a3, t0, t1, t2, t3, ws);

  finalize_kernel<<<1, 32, 0, stream>>>(ws, out);
}